// GCN_76201309766164
// MI455X (gfx1250) — compile-verified
//
#include <hip/hip_runtime.h>
#include <math.h>

// GCN (3-layer, 64-wide) for MI455X / gfx1250.
// Memory/atomic-bound workload -> fp32 everywhere; GEMM uses
// V_WMMA_F32_16X16X4_F32 (full-precision matrix pipe).

typedef __attribute__((ext_vector_type(2))) float v2f;
typedef __attribute__((ext_vector_type(8))) float v8f;

#define NF 64  // feature width for all layers

// ---------------- utility: zero a float buffer ----------------
__global__ void gcn_zero_f32(float* __restrict__ p, long n) {
    long i = (long)blockIdx.x * blockDim.x + threadIdx.x;
    long stride = (long)gridDim.x * blockDim.x;
    for (; i < n; i += stride) p[i] = 0.0f;
}

// ---------------- degree accumulation ----------------
__global__ void gcn_degree(const int* __restrict__ src, const int* __restrict__ dst,
                           float* __restrict__ deg_src, float* __restrict__ deg_dst, int E) {
    int e = blockIdx.x * blockDim.x + threadIdx.x;
    if (e < E) {
        atomicAdd(&deg_src[src[e]], 1.0f);
        atomicAdd(&deg_dst[dst[e]], 1.0f);
    }
}

// norm = rsqrt(max(deg, 1))   (in place, covers both buffers at once)
__global__ void gcn_norm(float* __restrict__ p, int n) {
    int i = blockIdx.x * blockDim.x + threadIdx.x;
    if (i < n) p[i] = rsqrtf(fmaxf(p[i], 1.0f));
}

// ---------------- edge gather + scatter-add ----------------
// 16 work-items per edge, one float4 feature chunk each.
__global__ __launch_bounds__(256) void gcn_edge_scatter(
    const float* __restrict__ x, const int* __restrict__ src, const int* __restrict__ dst,
    const float* __restrict__ norm_src, float* __restrict__ agg, int E) {
    long idx = (long)blockIdx.x * blockDim.x + threadIdx.x;
    long total = (long)E * 16;
    if (idx >= total) return;
    int e = (int)(idx >> 4);
    int c = (int)(idx & 15);
    int s = src[e];
    int d = dst[e];
    const float4* xs = (const float4*)(x + (size_t)s * NF);
    float4 v = xs[c];
    float sc = norm_src[s];
    float* ap = agg + (size_t)d * NF + c * 4;
    atomicAdd(ap + 0, v.x * sc);
    atomicAdd(ap + 1, v.y * sc);
    atomicAdd(ap + 2, v.z * sc);
    atomicAdd(ap + 3, v.w * sc);
}

// ---------------- WMMA GEMM: out = act(norm_dst .* agg @ W + b) ----------------
// One wave computes a 16-row x 64-col tile with 4 f32 16x16 accumulators.
// A layout (32-bit 16x4): lane L holds row M=L&15; VGPR0/1 hold K = k0 + 2*(L>>4) + {0,1}.
// B layout mirrors:       lane L holds col N = n0 + (L&15); same K pair.
// W is staged into LDS with K-pairs interleaved so each B operand is one
// aligned ds_load_b64:  Wl[(k>>1)*2*NF + col*2 + (k&1)].
// Bias is folded into the accumulator init (bias depends only on column).
// mode 0: ReLU ; mode 1: sigmoid + write thresholded clone.
__global__ __launch_bounds__(256) void gcn_gemm_wmma(
    const float* __restrict__ agg, const float* __restrict__ norm_dst,
    const float* __restrict__ W, const float* __restrict__ bias,
    float* __restrict__ out, float* __restrict__ clone_out, int N, int mode) {
    __shared__ float Wl[NF * NF];
    __shared__ float bl[NF];

    // cooperative stage of W (16 KB) into LDS, K-pair interleaved
    for (int i = threadIdx.x; i < (NF * NF) / 4; i += blockDim.x) {
        int k = i >> 4;              // source row (K)
        int c4 = (i & 15) * 4;       // source col group
        float4 w = ((const float4*)W)[i];
        int base = (k >> 1) * (2 * NF) + (k & 1);
        Wl[base + (c4 + 0) * 2] = w.x;
        Wl[base + (c4 + 1) * 2] = w.y;
        Wl[base + (c4 + 2) * 2] = w.z;
        Wl[base + (c4 + 3) * 2] = w.w;
    }
    if (threadIdx.x < NF) bl[threadIdx.x] = bias[threadIdx.x];
    __syncthreads();

    int wave = threadIdx.x >> 5;
    int lane = threadIdx.x & 31;
    int hi   = lane >> 4;   // 0: lanes 0-15, 1: lanes 16-31
    int l15  = lane & 15;

    int rowbase = blockIdx.x * 128 + wave * 16;  // 8 waves x 16 rows per block
    if (rowbase >= N) return;                    // wave-uniform: EXEC stays all-ones

    int row = rowbase + l15;                     // A-matrix row owned by this lane
    float nd = norm_dst[row];
    const v2f* af2 = (const v2f*)(agg + (size_t)row * NF);
    const v2f* Wp  = (const v2f*)Wl;             // [32 k-pairs][64 cols]

    // bias folded into C init: every C element this lane owns has col = t*16+l15
    float bv0 = bl[ 0 + l15], bv1 = bl[16 + l15], bv2 = bl[32 + l15], bv3 = bl[48 + l15];
    v8f acc0 = {bv0, bv0, bv0, bv0, bv0, bv0, bv0, bv0};
    v8f acc1 = {bv1, bv1, bv1, bv1, bv1, bv1, bv1, bv1};
    v8f acc2 = {bv2, bv2, bv2, bv2, bv2, bv2, bv2, bv2};
    v8f acc3 = {bv3, bv3, bv3, bv3, bv3, bv3, bv3, bv3};

#pragma unroll
    for (int k0 = 0; k0 < NF; k0 += 4) {
        v2f a = af2[(k0 >> 1) + hi];
        a.x *= nd;
        a.y *= nd;
        int kp = ((k0 >> 1) + hi) * NF;          // pair-row base in v2f units
        v2f b0 = Wp[kp +  0 + l15];
        v2f b1 = Wp[kp + 16 + l15];
        v2f b2 = Wp[kp + 32 + l15];
        v2f b3 = Wp[kp + 48 + l15];
        acc0 = __builtin_amdgcn_wmma_f32_16x16x4_f32(false, a, false, b0, (short)0, acc0, false, false);
        acc1 = __builtin_amdgcn_wmma_f32_16x16x4_f32(false, a, false, b1, (short)0, acc1, false, false);
        acc2 = __builtin_amdgcn_wmma_f32_16x16x4_f32(false, a, false, b2, (short)0, acc2, false, false);
        acc3 = __builtin_amdgcn_wmma_f32_16x16x4_f32(false, a, false, b3, (short)0, acc3, false, false);
    }

    // C/D layout: VGPR v -> row rowbase + v + 8*hi, col = tile*16 + l15
    v8f accs[4] = {acc0, acc1, acc2, acc3};
    if (mode == 0) {
#pragma unroll
        for (int t = 0; t < 4; t++) {
#pragma unroll
            for (int v = 0; v < 8; v++) {
                int r = rowbase + v + hi * 8;
                size_t o = (size_t)r * NF + t * 16 + l15;
                out[o] = fmaxf(accs[t][v], 0.0f);
            }
        }
    } else {
#pragma unroll
        for (int t = 0; t < 4; t++) {
#pragma unroll
            for (int v = 0; v < 8; v++) {
                int r = rowbase + v + hi * 8;
                size_t o = (size_t)r * NF + t * 16 + l15;
                float sg = 1.0f / (1.0f + __expf(-accs[t][v]));
                out[o] = sg;
                clone_out[o] = (sg >= 0.5f) ? 1.0f : 0.0f;
            }
        }
    }
}

// ---------------- driver ----------------
extern "C" void kernel_launch(void* const* d_in, const int* in_sizes, int n_in,
                              void* d_out, int out_size, void* d_ws, size_t ws_size,
                              hipStream_t stream) {
    const float* in_feat = (const float*)d_in[0];
    const int*   src     = (const int*)d_in[1];
    const int*   dst     = (const int*)d_in[2];
    const float* W1 = (const float*)d_in[3];
    const float* b1 = (const float*)d_in[4];
    const float* W2 = (const float*)d_in[5];
    const float* b2 = (const float*)d_in[6];
    const float* W3 = (const float*)d_in[7];
    const float* b3 = (const float*)d_in[8];

    const int N = in_sizes[0] / NF;   // 100000
    const int E = in_sizes[1];        // 1200000

    float* ws    = (float*)d_ws;
    float* nsrc  = ws;                          // [N]
    float* ndst  = ws + N;                      // [N]
    float* buf0  = ws + 2 * (size_t)N;          // [N*64]  agg
    float* buf1  = buf0 + (size_t)N * NF;       // [N*64]  h
    float* h_out     = (float*)d_out;           // [N*64]
    float* clone_out = h_out + (size_t)N * NF;  // [N*64]

    const long nfeat = (long)N * NF;
    const long ethreads = (long)E * 16;
    const int  eblocks  = (int)((ethreads + 255) / 256);
    const int  gblocks  = (N + 127) / 128;

    // degrees -> norms
    gcn_zero_f32<<<1024, 256, 0, stream>>>(nsrc, 2L * N);
    gcn_zero_f32<<<4096, 256, 0, stream>>>(buf0, nfeat);
    gcn_degree<<<(E + 255) / 256, 256, 0, stream>>>(src, dst, nsrc, ndst, E);
    gcn_norm<<<(2 * N + 255) / 256, 256, 0, stream>>>(ws, 2 * N);

    // layer 1: in_feat -> buf1 (ReLU)
    gcn_edge_scatter<<<eblocks, 256, 0, stream>>>(in_feat, src, dst, nsrc, buf0, E);
    gcn_gemm_wmma<<<gblocks, 256, 0, stream>>>(buf0, ndst, W1, b1, buf1, nullptr, N, 0);

    // layer 2: buf1 -> buf1 (ReLU)
    gcn_zero_f32<<<4096, 256, 0, stream>>>(buf0, nfeat);
    gcn_edge_scatter<<<eblocks, 256, 0, stream>>>(buf1, src, dst, nsrc, buf0, E);
    gcn_gemm_wmma<<<gblocks, 256, 0, stream>>>(buf0, ndst, W2, b2, buf1, nullptr, N, 0);

    // layer 3: buf1 -> d_out (sigmoid + threshold clone)
    gcn_zero_f32<<<4096, 256, 0, stream>>>(buf0, nfeat);
    gcn_edge_scatter<<<eblocks, 256, 0, stream>>>(buf1, src, dst, nsrc, buf0, E);
    gcn_gemm_wmma<<<gblocks, 256, 0, stream>>>(buf0, ndst, W3, b3, h_out, clone_out, N, 1);
}